// GCNLayer_11879879541106
// MI455X (gfx1250) — compile-verified
//
#include <hip/hip_runtime.h>

// SpMM segment-sum via V_WMMA_F32_16X16X4_F32 on gfx1250 (wave32).
// Memory-bound problem (~80 MB/call -> ~3.5us at 23.3 TB/s); WMMA is used to
// perform the in-tile segment reduction in fp32 and cut atomic traffic ~16x.

typedef __attribute__((ext_vector_type(2))) float v2f;
typedef __attribute__((ext_vector_type(8))) float v8f;

#define D_FEAT 64
#define TILE_E 16
#define WAVES_PER_BLOCK 8
#define BLOCKS 1024

__global__ __launch_bounds__(256) void spmm_wmma_kernel(
    const int* __restrict__ rows, const int* __restrict__ cols,
    const float* __restrict__ vals, const float* __restrict__ embeds,
    float* __restrict__ out, int numTiles, int numEdges)
{
    __shared__ int rowTab[WAVES_PER_BLOCK][16];

    const int lane  = threadIdx.x & 31;
    const int wid   = threadIdx.x >> 5;
    const int gwave = blockIdx.x * WAVES_PER_BLOCK + wid;
    const int nWaves = gridDim.x * WAVES_PER_BLOCK;
    const int half  = lane >> 4;      // 0: lanes 0-15, 1: lanes 16-31
    const int nl    = lane & 15;      // N (feature) index within 16-wide tile

    // Contiguous tile ranges per wave (uniform trip count within a wave;
    // keeps atomic targets spatially local since rows are sorted).
    const int tpw = (numTiles + nWaves - 1) / nWaves;
    int t0 = gwave * tpw;
    int t1 = t0 + tpw; if (t1 > numTiles) t1 = numTiles;

    for (int t = t0; t < t1; ++t) {
        const int e0 = t * TILE_E;

        // Lanes 0..15 hold one edge each of this tile.
        int r = 0, c = 0; float v = 0.f;
        if (lane < 16) {
            r = rows[e0 + lane];
            c = cols[e0 + lane];
            v = vals[e0 + lane];
        }

        // Local row rank within the sorted tile: prefix count of boundaries.
        int rup   = __shfl_up(r, 1);
        int flag  = (lane > 0 && lane < 16 && r != rup) ? 1 : 0;
        unsigned mask = (unsigned)__ballot(flag);          // wave32: low 32 bits
        int local  = __popc(mask & ((2u << lane) - 1));    // inclusive prefix
        int nLocal = __shfl(local, 15) + 1;                // distinct rows, 1..16

        // local -> global row table (one writer per local index; same-wave
        // LDS ops are in order; wave_barrier stops compiler reordering).
        if (lane < 16 && (flag || lane == 0)) rowTab[wid][local] = r;
        __builtin_amdgcn_wave_barrier();

        // Accumulators: 4 feature tiles of 16 columns each (16x16 f32 D).
        v8f acc[4] = {v8f{}, v8f{}, v8f{}, v8f{}};

        // K runs over the 16 edges in chunks of 4 (WMMA K=4).
        // A 16x4 f32 layout: lane=M; VGPR0/1 = K {k0,k0+1} (lanes 0-15)
        //                                       K {k0+2,k0+3} (lanes 16-31).
        // B 4x16 layout mirrors the half-wave K split with lane(&15)=N.
#pragma unroll
        for (int k0 = 0; k0 < 16; k0 += 4) {
            const int ka0 = k0 + 2 * half;
            const int ka1 = ka0 + 1;
            // Broadcast edge data (held in lanes 0..15) — all lanes active.
            int   l0 = __shfl(local, ka0);
            int   l1 = __shfl(local, ka1);
            float s0 = __shfl(v, ka0);
            float s1 = __shfl(v, ka1);
            int   c0 = __shfl(c, ka0);
            int   c1 = __shfl(c, ka1);

            v2f a;
            a.x = (l0 == nl) ? s0 : 0.f;   // S[m][ka0], m = nl
            a.y = (l1 == nl) ? s1 : 0.f;   // S[m][ka1]

            const float* g0 = embeds + (long)c0 * D_FEAT + nl;
            const float* g1 = embeds + (long)c1 * D_FEAT + nl;
#pragma unroll
            for (int ft = 0; ft < 4; ++ft) {
                v2f b;
                b.x = g0[ft * 16];         // B[ka0][n], coalesced per half-wave
                b.y = g1[ft * 16];         // B[ka1][n]
                acc[ft] = __builtin_amdgcn_wmma_f32_16x16x4_f32(
                    false, a, false, b, (short)0, acc[ft], false, false);
            }
        }

        // Scatter: D VGPR j holds row j (lanes 0-15) / row j+8 (lanes 16-31),
        // column nl. Only the nLocal populated rows are written.
#pragma unroll
        for (int ft = 0; ft < 4; ++ft) {
            const int fb = ft * 16;
#pragma unroll
            for (int k = 0; k < 8; ++k) {
                int m = half * 8 + k;
                int grow = rowTab[wid][m];       // read by all lanes (valid addr)
                if (m < nLocal)
                    atomicAdd(&out[(long)grow * D_FEAT + fb + nl], acc[ft][k]);
            }
        }
        __builtin_amdgcn_wave_barrier();   // order LDS reads vs next tile writes
    }

    // Tail edges (E % 16) handled scalar by the first wave (0 for E=1.6M).
    if (gwave == 0) {
        for (int e = numTiles * TILE_E; e < numEdges; ++e) {
            int   rr = rows[e];
            int   cc = cols[e];
            float vv = vals[e];
#pragma unroll
            for (int j = 0; j < 2; ++j) {
                int f = lane + j * 32;
                atomicAdd(&out[(long)rr * D_FEAT + f],
                          vv * embeds[(long)cc * D_FEAT + f]);
            }
        }
    }
}

extern "C" void kernel_launch(void* const* d_in, const int* in_sizes, int n_in,
                              void* d_out, int out_size, void* d_ws, size_t ws_size,
                              hipStream_t stream) {
    const int*   rows   = (const int*)d_in[0];
    const int*   cols   = (const int*)d_in[1];
    const float* vals   = (const float*)d_in[2];
    const float* embeds = (const float*)d_in[3];
    float*       out    = (float*)d_out;

    const int E = in_sizes[0];
    const int numTiles = E / TILE_E;

    // Atomic accumulation: output must be zeroed every call (capturable).
    hipMemsetAsync(d_out, 0, (size_t)out_size * sizeof(float), stream);

    spmm_wmma_kernel<<<BLOCKS, WAVES_PER_BLOCK * 32, 0, stream>>>(
        rows, cols, vals, embeds, out, numTiles, E);
}